// WeightedSum_86535001080433
// MI455X (gfx1250) — compile-verified
//
#include <hip/hip_runtime.h>

// Problem constants (from reference): B=64, F=128, CI=64, CO=64, R=4
#define BB 64
#define FF 128
#define CIN 64
#define CON 64
#define RR 4

typedef float v2f __attribute__((ext_vector_type(2)));
typedef float v8f __attribute__((ext_vector_type(8)));

// LDS layout (float offsets). Stride 65 padding -> bank-conflict-free column access.
#define E_STRIDE 65
#define E_R (BB * E_STRIDE)          // 4160 floats per r-plane of E
#define W_STRIDE 65
#define W_R (CON * W_STRIDE)         // 4160 floats per r-plane of Wt
#define LE_OFF 0
#define LW_OFF (RR * E_R)            // 16640
#define LM_OFF (LW_OFF + RR * W_R)   // 33280
#define LDS_FLOATS (LM_OFF + RR * BB)  // 33536
#define LDS_BYTES (LDS_FLOATS * 4)     // 134144 bytes (< 320KB WGP LDS)

extern "C" __global__ __launch_bounds__(256)
void weightedsum_lse_wmma(const float* __restrict__ ll,
                          const float* __restrict__ wgt,
                          float* __restrict__ out) {
    extern __shared__ float smem[];
    const int f   = blockIdx.x;        // one workgroup per feature f
    const int tid = threadIdx.x;

    // ---------------- Phase 0: coalesced float4 staging (r=0..3 contiguous) ----------
    const float4* ll4 = (const float4*)ll;   // ll[b,f,ci,0:4] at flat (b*FF+f)*CIN+ci
    const float4* w4  = (const float4*)wgt;  // w[f,ci,co,0:4] at flat (f*CIN+ci)*CON+co
    #pragma unroll
    for (int i = 0; i < 16; ++i) {
        const int idx = tid + i * 256;       // 0..4095
        const int hi = idx >> 6, lo = idx & 63;

        // ll tile: hi=b, lo=ci  -> E[r][b][ci] (raw for now)
        float4 v = ll4[(hi * FF + f) * CIN + lo];
        smem[LE_OFF + 0 * E_R + hi * E_STRIDE + lo] = v.x;
        smem[LE_OFF + 1 * E_R + hi * E_STRIDE + lo] = v.y;
        smem[LE_OFF + 2 * E_R + hi * E_STRIDE + lo] = v.z;
        smem[LE_OFF + 3 * E_R + hi * E_STRIDE + lo] = v.w;

        // weight tile: hi=ci, lo=co -> transposed Wt[r][co][ci]
        float4 wv = w4[(f * CIN + hi) * CON + lo];
        smem[LW_OFF + 0 * W_R + lo * W_STRIDE + hi] = wv.x;
        smem[LW_OFF + 1 * W_R + lo * W_STRIDE + hi] = wv.y;
        smem[LW_OFF + 2 * W_R + lo * W_STRIDE + hi] = wv.z;
        smem[LW_OFF + 3 * W_R + lo * W_STRIDE + hi] = wv.w;
    }
    __syncthreads();

    // ---------------- Phase 1: per-(b,r) row max over ci, then exp(ll - m) ----------
    {
        const int b = tid & 63;
        const int r = tid >> 6;
        float* row = &smem[LE_OFF + r * E_R + b * E_STRIDE];
        float m = row[0];
        #pragma unroll 8
        for (int ci = 1; ci < CIN; ++ci) m = fmaxf(m, row[ci]);
        smem[LM_OFF + r * BB + b] = m;
        #pragma unroll 8
        for (int ci = 0; ci < CIN; ++ci) row[ci] = __expf(row[ci] - m);
    }
    __syncthreads();

    // ---------------- Phase 2: WMMA f32 16x16x4 GEMMs: S_r = E_r (64x64) * W_r (64x64)
    const int wave  = tid >> 5;
    const int lane  = tid & 31;
    const int r     = wave >> 1;       // 2 waves per r
    const int mhalf = wave & 1;        // rows 0-31 or 32-63
    const int lrow  = lane & 15;
    const int koff  = (lane >> 4) << 1;  // lanes 16-31 hold K+2 / K+3

    const float* Eb = &smem[LE_OFF + r * E_R];
    const float* Wb = &smem[LW_OFF + r * W_R];

    v8f acc[8];
    #pragma unroll
    for (int mt = 0; mt < 2; ++mt) {
        const int arow = (mhalf * 2 + mt) * 16 + lrow;
        #pragma unroll
        for (int nt = 0; nt < 4; ++nt) {
            const int bcol = nt * 16 + lrow;
            v8f c = {0.f, 0.f, 0.f, 0.f, 0.f, 0.f, 0.f, 0.f};
            #pragma unroll
            for (int k = 0; k < CIN; k += 4) {
                const int kk = k + koff;
                v2f a, b;
                a.x = Eb[arow * E_STRIDE + kk];
                a.y = Eb[arow * E_STRIDE + kk + 1];
                b.x = Wb[bcol * W_STRIDE + kk];
                b.y = Wb[bcol * W_STRIDE + kk + 1];
                c = __builtin_amdgcn_wmma_f32_16x16x4_f32(
                        /*neg_a=*/false, a, /*neg_b=*/false, b,
                        /*c_mod=*/(short)0, c, /*reuse_a=*/false, /*reuse_b=*/false);
            }
            acc[mt * 4 + nt] = c;
        }
    }
    __syncthreads();   // all waves done reading E/W; safe to reuse E region as output S

    // ---------------- Phase 3a: accumulators -> LDS  S[r][b][co] (aliases E region) --
    {
        float* O = &smem[LE_OFF + r * E_R];
        const int rowadd = (lane >> 4) << 3;   // VGPR v: lanes0-15 -> M=v, lanes16-31 -> M=v+8
        #pragma unroll
        for (int mt = 0; mt < 2; ++mt) {
            const int base_row = (mhalf * 2 + mt) * 16 + rowadd;
            #pragma unroll
            for (int nt = 0; nt < 4; ++nt) {
                const int col = nt * 16 + lrow;
                v8f c = acc[mt * 4 + nt];
                #pragma unroll
                for (int v = 0; v < 8; ++v)
                    O[(base_row + v) * E_STRIDE + col] = c[v];
            }
        }
    }
    __syncthreads();

    // ---------------- Phase 3b: out = m + log(S), coalesced float4 stores ------------
    float4* out4 = (float4*)out;   // out[b,f,co,0:4] at flat (b*FF+f)*CON+co
    #pragma unroll
    for (int i = 0; i < 16; ++i) {
        const int idx = tid + i * 256;
        const int b = idx >> 6, co = idx & 63;
        float4 o;
        o.x = smem[LM_OFF + 0 * BB + b] + __logf(smem[LE_OFF + 0 * E_R + b * E_STRIDE + co]);
        o.y = smem[LM_OFF + 1 * BB + b] + __logf(smem[LE_OFF + 1 * E_R + b * E_STRIDE + co]);
        o.z = smem[LM_OFF + 2 * BB + b] + __logf(smem[LE_OFF + 2 * E_R + b * E_STRIDE + co]);
        o.w = smem[LM_OFF + 3 * BB + b] + __logf(smem[LE_OFF + 3 * E_R + b * E_STRIDE + co]);
        out4[(b * FF + f) * CON + co] = o;
    }
}

extern "C" void kernel_launch(void* const* d_in, const int* in_sizes, int n_in,
                              void* d_out, int out_size, void* d_ws, size_t ws_size,
                              hipStream_t stream) {
    (void)in_sizes; (void)n_in; (void)d_ws; (void)ws_size; (void)out_size;
    const float* ll  = (const float*)d_in[0];   // (B,F,CI,R) f32
    const float* wgt = (const float*)d_in[1];   // (F,CI,CO,R) f32
    float* out = (float*)d_out;                 // (B,F,CO,R) f32

    weightedsum_lse_wmma<<<dim3(FF), dim3(256), LDS_BYTES, stream>>>(ll, wgt, out);
}